// Toy_20615843020916
// MI455X (gfx1250) — compile-verified
//
#include <hip/hip_runtime.h>

// CDNA5 (gfx1250) wave32 WMMA implementation of the toy-MLP force kernel.
// force(p) = -d/dp [ sum( W3 @ relu(W2 @ relu(W1 p + b1) + b2) ) ]
//
// Strategy: batch 16 positions per WMMA tile (positions = N dim / lanes),
// weights live in A operands (VGPR0 = {K0 | K2}, VGPR1 = {K1 | K3} across the
// two lane halves, per the ISA 32-bit A layout), biases folded into C
// (C[m][n] = b[m] in the documented C/D layout). Layer chaining uses
// ds_bpermute + cndmask to re-form D rows into the next B operand.
//
// Persistent waves: weight-operand construction hoisted out of a wave-uniform
// grid-stride loop. Main loop handles only full 32-position chunks with
// unconditional load/store; a single clamped tail step covers N % 32.
// ReLU sign bits (d1 > 0) are computed once and reused for the backward mask.

typedef __attribute__((ext_vector_type(2))) float v2f;
typedef __attribute__((ext_vector_type(8))) float v8f;

__device__ __forceinline__ float bperm(int idx_bytes, float v) {
    return __int_as_float(__builtin_amdgcn_ds_bpermute(idx_bytes, __float_as_int(v)));
}

__device__ __forceinline__ v8f wmma4(v2f a, v2f b, v8f c) {
    // (neg_a, A, neg_b, B, c_mod, C, reuse_a, reuse_b)
    return __builtin_amdgcn_wmma_f32_16x16x4_f32(false, a, false, b, (short)0, c,
                                                 false, false);
}

__global__ __launch_bounds__(256) void toy_force_kernel(
    const float* __restrict__ pos,
    const float* __restrict__ W1, const float* __restrict__ b1,
    const float* __restrict__ W2, const float* __restrict__ b2,
    const float* __restrict__ W3,
    float* __restrict__ out, int N)
{
    const unsigned tid   = threadIdx.x;
    const unsigned lane  = tid & 31u;
    const unsigned wave  = tid >> 5;
    const unsigned waveG = blockIdx.x * 8u + wave;           // global wave id
    const unsigned waveT = gridDim.x * 8u;                   // total waves

    const bool     lo    = lane < 16u;
    const unsigned m     = lane & 15u;
    const int      pLow  = (int)(m << 2);          // bpermute: fetch from lane m
    const int      pHigh = (int)((m | 16u) << 2);  // bpermute: fetch from lane m+16

    // ---------------- per-wave constant WMMA operands (hoisted) ----------------
    // A1 = W1 padded to 16x4 (rows 0..7 valid, K=0..1 valid)
    const unsigned m8 = m & 7u, m4 = m & 3u, m2 = m & 1u;
    float w1c0 = W1[2u*m8 + 0u], w1c1 = W1[2u*m8 + 1u];
    v2f A1; A1.x = (lo && m < 8u) ? w1c0 : 0.0f;   // K0 | K2(=0)
            A1.y = (lo && m < 8u) ? w1c1 : 0.0f;   // K1 | K3(=0)

    // A2a = W2[:,0:4], A2b = W2[:,4:8], rows 0..3 valid
    float a2a0 = W2[8u*m4 + (lo ? 0u : 2u)];
    float a2a1 = W2[8u*m4 + (lo ? 1u : 3u)];
    float a2b0 = W2[8u*m4 + 4u + (lo ? 0u : 2u)];
    float a2b1 = W2[8u*m4 + 4u + (lo ? 1u : 3u)];
    v2f A2a; A2a.x = (m < 4u) ? a2a0 : 0.0f; A2a.y = (m < 4u) ? a2a1 : 0.0f;
    v2f A2b; A2b.x = (m < 4u) ? a2b0 : 0.0f; A2b.y = (m < 4u) ? a2b1 : 0.0f;

    // Ag = W2^T (8x4): A[m][k] = W2[k][m]
    float ag0 = W2[8u*(lo ? 0u : 2u) + m8];
    float ag1 = W2[8u*(lo ? 1u : 3u) + m8];
    v2f Ag; Ag.x = (m < 8u) ? ag0 : 0.0f; Ag.y = (m < 8u) ? ag1 : 0.0f;

    // Af = -W1^T (2x8) split into K=0..3 / K=4..7 chunks: A[m][k] = -W1[2k+m]
    float af0 = -W1[2u*(lo ? 0u : 2u) + m2];
    float af1 = -W1[2u*(lo ? 1u : 3u) + m2];
    float af4 = -W1[2u*((lo ? 0u : 2u) + 4u) + m2];
    float af5 = -W1[2u*((lo ? 1u : 3u) + 4u) + m2];
    v2f Afa; Afa.x = (m < 2u) ? af0 : 0.0f; Afa.y = (m < 2u) ? af1 : 0.0f;
    v2f Afb; Afb.x = (m < 2u) ? af4 : 0.0f; Afb.y = (m < 2u) ? af5 : 0.0f;

    // Bias accumulators: C[v] holds row v (lanes 0-15) / row v+8 (lanes 16-31)
    v8f C1;
    #pragma unroll
    for (int v = 0; v < 8; ++v) C1[v] = lo ? b1[v] : 0.0f;   // rows 8..15 -> 0
    v8f C2;
    #pragma unroll
    for (int v = 0; v < 4; ++v) C2[v] = lo ? b2[v] : 0.0f;
    #pragma unroll
    for (int v = 4; v < 8; ++v) C2[v] = 0.0f;

    // s3[j] = W3[0,j] + W3[1,j] (grad of sum wrt h2), masked to valid lane half
    float s3sel[4];
    #pragma unroll
    for (int j = 0; j < 4; ++j) s3sel[j] = lo ? (W3[j] + W3[4u + j]) : 0.0f;

    const v8f Zero = {};

    auto tile = [&](v2f Bp, float& fx, float& fy) {
        // layer 1: h1 = relu(W1 p + b1)    (rows 0..7)
        v8f d1 = wmma4(A1, Bp, C1);
        bool gt1[8];
        v8f h1;
        #pragma unroll
        for (int v = 0; v < 8; ++v) {
            gt1[v] = d1[v] > 0.0f;               // one cmp: reused fwd + bwd
            h1[v]  = gt1[v] ? d1[v] : 0.0f;
        }

        // re-form h1 rows into B operands (rows 0-3 and 4-7)
        float t2 = bperm(pLow, h1[2]), t3 = bperm(pLow, h1[3]);
        float t6 = bperm(pLow, h1[6]), t7 = bperm(pLow, h1[7]);
        v2f Ba; Ba.x = lo ? h1[0] : t2; Ba.y = lo ? h1[1] : t3;
        v2f Bb; Bb.x = lo ? h1[4] : t6; Bb.y = lo ? h1[5] : t7;

        // layer 2 (K=8 split): h2pre = W2 h1 + b2   (rows 0..3)
        v8f d2 = wmma4(A2b, Bb, C2);
        d2     = wmma4(A2a, Ba, d2);

        // g2 = s3 * 1[h2pre > 0]
        float g2[4];
        #pragma unroll
        for (int v = 0; v < 4; ++v) g2[v] = (d2[v] > 0.0f) ? s3sel[v] : 0.0f;
        float u2 = bperm(pLow, g2[2]), u3 = bperm(pLow, g2[3]);
        v2f Bg; Bg.x = lo ? g2[0] : u2; Bg.y = lo ? g2[1] : u3;

        // g1 = (W2^T g2) * 1[h1 > 0]   (reuse gt1; padded rows have gt1=false)
        v8f dg = wmma4(Ag, Bg, Zero);
        v8f g1;
        #pragma unroll
        for (int v = 0; v < 8; ++v) g1[v] = gt1[v] ? dg[v] : 0.0f;

        float r2 = bperm(pLow, g1[2]), r3 = bperm(pLow, g1[3]);
        float r6 = bperm(pLow, g1[6]), r7 = bperm(pLow, g1[7]);
        v2f Bga; Bga.x = lo ? g1[0] : r2; Bga.y = lo ? g1[1] : r3;
        v2f Bgb; Bgb.x = lo ? g1[4] : r6; Bgb.y = lo ? g1[5] : r7;

        // force = (-W1^T) g1   (K=8 split; rows 0..1)
        v8f df = wmma4(Afb, Bgb, Zero);
        df     = wmma4(Afa, Bga, df);
        fx = df[0]; fy = df[1];
    };

    // process one 32-position chunk; `tail` folds away after inlining
    auto process = [&](unsigned base, bool tail) {
        const unsigned idx  = base + lane;
        const unsigned idxc = (!tail) ? idx
                              : ((idx < (unsigned)N) ? idx : (unsigned)(N - 1));
        v2f p = ((const v2f*)pos)[idxc];

        // Tile 0: columns = positions of lanes 0..15 ; Tile 1: of lanes 16..31
        float xh = bperm(pHigh, p.x);
        float yh = bperm(pHigh, p.y);
        v2f Bp0; Bp0.x = lo ? p.x : 0.0f; Bp0.y = lo ? p.y : 0.0f;  // K2/K3 pad = 0
        v2f Bp1; Bp1.x = lo ? xh  : 0.0f; Bp1.y = lo ? yh  : 0.0f;

        float fx0, fy0, fx1, fy1;
        tile(Bp0, fx0, fy0);
        tile(Bp1, fx1, fy1);

        // lane n<16 stores tile0 column n ; lane n>=16 stores tile1 column n-16
        float fxh = bperm(pLow, fx1);
        float fyh = bperm(pLow, fy1);
        v2f f; f.x = lo ? fx0 : fxh; f.y = lo ? fy0 : fyh;

        if (!tail || idx < (unsigned)N) ((v2f*)out)[idx] = f;
    };

    // ---------- wave-uniform grid-stride loop over full chunks ----------
    const unsigned stride = waveT * 32u;
    unsigned base = waveG * 32u;
    for (; base + 32u <= (unsigned)N; base += stride)
        process(base, false);                    // no clamps, no predication
    if (base < (unsigned)N)
        process(base, true);                     // clamped tail (N % 32)
}

extern "C" void kernel_launch(void* const* d_in, const int* in_sizes, int n_in,
                              void* d_out, int out_size, void* d_ws, size_t ws_size,
                              hipStream_t stream) {
    const float* pos = (const float*)d_in[0];
    const float* W1  = (const float*)d_in[1];
    const float* b1  = (const float*)d_in[2];
    const float* W2  = (const float*)d_in[3];
    const float* b2  = (const float*)d_in[4];
    const float* W3  = (const float*)d_in[5];
    float* out = (float*)d_out;

    const int N = in_sizes[0] / 2;                 // number of 2-D positions
    // Persistent-ish grid: enough waves for full occupancy, one chunk each until the 2048-block cap.
    int blocks = (N + 255) / 256;                  // 8 waves/block * 32 pos/wave
    if (blocks > 2048) blocks = 2048;
    toy_force_kernel<<<blocks, 256, 0, stream>>>(pos, W1, b1, W2, b2, W3, out, N);
}